// NonsatActivation_17308718203490
// MI455X (gfx1250) — compile-verified
//
#include <hip/hip_runtime.h>

// NonsatActivation: elementwise Newton solve of y^3/3 + y = x over fp32.
// Memory-bound streaming map: 1 GiB HBM traffic @ 23.3 TB/s => ~46 us floor.
// Strategy: 2x unrolled grid-stride => two 128-bit non-temporal loads in
// flight per wave (MLP) and 8 independent Newton chains per thread (ILP) so
// the TRANS-pipe v_rcp_f32 latency never stalls the wave; NT stores since the
// 1 GiB stream has zero reuse (>> 192MB L2); wave32 256-thread blocks.

typedef float v4f __attribute__((ext_vector_type(4)));

#define NONSAT_EP      1e-4f
#define NONSAT_MAXIT   100

// One Newton step: y' = (2/3 * y^3 + x) / (y^2 + 1).
// Denominator >= 1, so the fast reciprocal is safe (no denorm/overflow).
__device__ __forceinline__ float nonsat_step(float y, float x) {
    float y2  = y * y;
    float y3  = y2 * y;
    float num = __builtin_fmaf(0x1.555556p-1f, y3, x);   // (2/3)*y^3 + x
    float den = y2 + 1.0f;
    return num * __builtin_amdgcn_rcpf(den);             // TRANS v_rcp_f32
}

__device__ __forceinline__ v4f nonsat_step4(v4f y, v4f x) {
    v4f r;
    r.x = nonsat_step(y.x, x.x);
    r.y = nonsat_step(y.y, x.y);
    r.z = nonsat_step(y.z, x.z);
    r.w = nonsat_step(y.w, x.w);
    return r;
}

__device__ __forceinline__ float maxabs4(v4f a, v4f b) {
    float m0 = __builtin_fmaxf(__builtin_fabsf(a.x - b.x),
                               __builtin_fabsf(a.y - b.y));
    float m1 = __builtin_fmaxf(__builtin_fabsf(a.z - b.z),
                               __builtin_fabsf(a.w - b.w));
    return __builtin_fmaxf(m0, m1);
}

// Scalar path for the (tiny) tail.
__device__ __forceinline__ float nonsat_one(float x) {
    float y  = x;
    float yn = nonsat_step(y, x);
#pragma unroll 1
    for (int it = 0; it <= NONSAT_MAXIT; ++it) {
        if (__builtin_fabsf(yn - y) <= NONSAT_EP) break;
        y  = yn;
        yn = nonsat_step(y, x);
    }
    return yn;   // == step(y_prev, x) in the reference
}

__global__ __launch_bounds__(256)
void NonsatActivation_kernel(const float* __restrict__ in,
                             float* __restrict__ out,
                             long long n) {
    const long long n4      = n >> 2;
    const long long stride  = (long long)gridDim.x * blockDim.x;
    const long long stride2 = stride * 2;
    const long long tid     = (long long)blockIdx.x * blockDim.x + threadIdx.x;

    const v4f* __restrict__ in4  = (const v4f*)in;
    v4f* __restrict__       out4 = (v4f*)out;

    for (long long i = tid; i < n4; i += stride2) {
        const long long j   = i + stride;
        const bool      two = (j < n4);

        // Two independent 128-bit NT loads in flight (memory-level parallelism).
        v4f x0 = __builtin_nontemporal_load(in4 + i);
        v4f x1 = two ? __builtin_nontemporal_load(in4 + j) : x0;

        // Eight independent Newton chains -> deep ILP hides the serial
        // dependency of the iteration and the TRANS rcp latency.
        v4f y0  = x0,               y1  = x1;
        v4f yn0 = nonsat_step4(y0, x0);
        v4f yn1 = nonsat_step4(y1, x1);
#pragma unroll 1
        for (int it = 0; it <= NONSAT_MAXIT; ++it) {
            float m = __builtin_fmaxf(maxabs4(yn0, y0), maxabs4(yn1, y1));
            if (m <= NONSAT_EP) break;   // converged chains sit at the fixed
            y0  = yn0;                   // point; extra steps are no-ops
            y1  = yn1;
            yn0 = nonsat_step4(y0, x0);
            yn1 = nonsat_step4(y1, x1);
        }

        __builtin_nontemporal_store(yn0, out4 + i);      // global_store_b128 NT
        if (two) __builtin_nontemporal_store(yn1, out4 + j);
    }

    // Scalar tail (n % 4 elements); any grid covers it since tail < 4.
    const long long base = n4 << 2;
    if (base + tid < n) {
        out[base + tid] = nonsat_one(in[base + tid]);
    }
}

extern "C" void kernel_launch(void* const* d_in, const int* in_sizes, int n_in,
                              void* d_out, int out_size, void* d_ws, size_t ws_size,
                              hipStream_t stream) {
    (void)n_in; (void)d_ws; (void)ws_size; (void)out_size;

    const float* x   = (const float*)d_in[0];
    float*       out = (float*)d_out;
    const long long n = (long long)in_sizes[0];   // 32*4096*1024 = 134,217,728

    const int  block  = 256;                      // 8 wave32 waves per block
    long long  chunks = (n >> 3) / block + 1;     // 8 floats per thread-trip
    int grid = (int)(chunks < 32768 ? (chunks > 0 ? chunks : 1) : 32768);

    NonsatActivation_kernel<<<grid, block, 0, stream>>>(x, out, n);
}